// H2GCNEncoder_88304527606668
// MI455X (gfx1250) — compile-verified
//
#include <hip/hip_runtime.h>

typedef __attribute__((ext_vector_type(2))) float v2f;
typedef __attribute__((ext_vector_type(8))) float v8f;

#define LDS_STRIDE 132   // 128 + 4 pad: lane m -> bank 4m (+0..3), conflict-free for wave32

// ---------------------------------------------------------------------------
// Degree / utility kernels
// ---------------------------------------------------------------------------
__global__ void fill_ones_kernel(float* __restrict__ p, int n) {
    int i = blockIdx.x * blockDim.x + threadIdx.x;
    if (i < n) p[i] = 1.0f;   // self-loop contributes 1 to every degree
}

__global__ void deg_edges_kernel(const int* __restrict__ row, float* __restrict__ deg, int E) {
    int e = blockIdx.x * blockDim.x + threadIdx.x;
    if (e < E) {
        __hip_atomic_fetch_add(&deg[row[e]], 1.0f,
                               __ATOMIC_RELAXED, __HIP_MEMORY_SCOPE_AGENT);
    }
}

__global__ void invert_deg_kernel(float* __restrict__ deg, int n) {
    int i = blockIdx.x * blockDim.x + threadIdx.x;
    if (i < n) deg[i] = 1.0f / fmaxf(deg[i], 1.0f);
}

// agg = src  (self-loop term of the mean aggregation), float4 vectorized
__global__ void copy_f4_kernel(const float* __restrict__ src, float* __restrict__ dst, int n4) {
    int i = blockIdx.x * blockDim.x + threadIdx.x;
    if (i < n4) {
        ((float4*)dst)[i] = ((const float4*)src)[i];
    }
}

// ---------------------------------------------------------------------------
// Edge scatter-add: one wave (32 lanes) per edge, 4 channels per lane.
// row/col loads are wave-uniform -> scalar loads; feature gather is a
// coalesced float4; accumulation uses hardware global_atomic_add_f32.
// ---------------------------------------------------------------------------
__global__ __launch_bounds__(256)
void scatter_edges_kernel(const float* __restrict__ feat, float* __restrict__ acc,
                          const int* __restrict__ row, const int* __restrict__ col, int E) {
    int t = blockIdx.x * blockDim.x + threadIdx.x;
    int e = t >> 5;
    if (e >= E) return;
    int ch = (t & 31) * 4;
    int r = row[e];
    int c = col[e];
    float4 v = *(const float4*)(feat + (size_t)c * 128 + ch);
    float* dst = acc + (size_t)r * 128 + ch;
    __hip_atomic_fetch_add(dst + 0, v.x, __ATOMIC_RELAXED, __HIP_MEMORY_SCOPE_AGENT);
    __hip_atomic_fetch_add(dst + 1, v.y, __ATOMIC_RELAXED, __HIP_MEMORY_SCOPE_AGENT);
    __hip_atomic_fetch_add(dst + 2, v.z, __ATOMIC_RELAXED, __HIP_MEMORY_SCOPE_AGENT);
    __hip_atomic_fetch_add(dst + 3, v.w, __ATOMIC_RELAXED, __HIP_MEMORY_SCOPE_AGENT);
}

// ---------------------------------------------------------------------------
// WMMA fp32 GEMM:  out[N,128] = epilogue( A[N,128] @ B[128,128] (+ A2 @ B2) )
// epilogue: optional per-row scale (1/deg), +bias, optional relu, optional
// second output (write-through init for the next hop's aggregation buffer).
// Block: 256 threads = 8 waves; block computes a 16x128 output tile.
// A tile staged in LDS (padded), each wave owns one 16-wide N slice of B.
// Uses V_WMMA_F32_16X16X4_F32, 32 unrolled K-steps per 128-deep pass.
// ---------------------------------------------------------------------------
__device__ __forceinline__
void load_tile(const float* __restrict__ A, float* As, int m0, int N, int tid) {
#pragma unroll
    for (int i = 0; i < 2; ++i) {
        int e = tid * 4 + i * 1024;          // 0..2047, float4-granular
        int r = e >> 7;                      // row 0..15
        int c = e & 127;                     // col, multiple of 4
        int m = m0 + r;
        if (m >= N) m = N - 1;               // tail clamp (harmless duplicate)
        float4 v = *(const float4*)(A + (size_t)m * 128 + c);
        *(float4*)&As[r * LDS_STRIDE + c] = v;
    }
}

__device__ __forceinline__
v8f kloop_128(const float* As, const float* __restrict__ Bp, v8f acc, int nlo, int khalf) {
    // A lane layout (16x4 f32): lanes 0-15 -> K = ks+0/ks+1, lanes 16-31 -> K = ks+2/ks+3
    const float* ap = As + nlo * LDS_STRIDE + 2 * khalf;
#pragma unroll
    for (int ks = 0; ks < 128; ks += 4) {
        v2f a = *(const v2f*)(ap + ks);      // dual 64-bit ds load (merged by compiler)
        v2f b;
        b.x = Bp[(size_t)(ks + 2 * khalf + 0) * 128];
        b.y = Bp[(size_t)(ks + 2 * khalf + 1) * 128];
        acc = __builtin_amdgcn_wmma_f32_16x16x4_f32(
            /*neg_a=*/false, a, /*neg_b=*/false, b,
            /*c_mod=*/(short)0, acc, /*reuse_a=*/false, /*reuse_b=*/false);
    }
    return acc;
}

__global__ __launch_bounds__(256)
void gemm_wmma_kernel(const float* __restrict__ A,  const float* __restrict__ B,
                      const float* __restrict__ A2, const float* __restrict__ B2,
                      const float* __restrict__ bias, const float* __restrict__ scale,
                      float* __restrict__ out, float* __restrict__ out2,
                      int N, int relu) {
    __shared__ float As[16 * LDS_STRIDE];
    const int tid   = threadIdx.x;
    const int lane  = tid & 31;
    const int wave  = tid >> 5;              // 0..7 -> N tile
    const int m0    = blockIdx.x * 16;       // M tile base
    const int nlo   = lane & 15;
    const int khalf = lane >> 4;             // 0 or 1
    const int n0    = wave * 16 + nlo;       // output column

    load_tile(A, As, m0, N, tid);
    __syncthreads();

    v8f acc = {0.f, 0.f, 0.f, 0.f, 0.f, 0.f, 0.f, 0.f};
    acc = kloop_128(As, B + n0, acc, nlo, khalf);

    if (A2 != nullptr) {                     // concat layer: second K=128 pass
        __syncthreads();
        load_tile(A2, As, m0, N, tid);
        __syncthreads();
        acc = kloop_128(As, B2 + n0, acc, nlo, khalf);
    }

    const float bv = bias[n0];
#pragma unroll
    for (int v = 0; v < 8; ++v) {
        int m = m0 + v + 8 * khalf;          // C/D layout: VGPR v -> rows v / v+8
        if (m >= N) continue;
        float val = acc[v];
        if (scale) val *= scale[m];          // D^-1 (S @ W) == (D^-1 S) @ W
        val += bv;
        if (relu) val = fmaxf(val, 0.0f);
        out[(size_t)m * 128 + n0] = val;
        if (out2) out2[(size_t)m * 128 + n0] = val;  // fused init of next agg buffer
    }
}

// ---------------------------------------------------------------------------
// Host-side orchestration
// ---------------------------------------------------------------------------
extern "C" void kernel_launch(void* const* d_in, const int* in_sizes, int n_in,
                              void* d_out, int out_size, void* d_ws, size_t ws_size,
                              hipStream_t stream) {
    const float* x    = (const float*)d_in[0];
    const int*   ei   = (const int*)  d_in[1];
    const float* W1   = (const float*)d_in[2];
    const float* b1   = (const float*)d_in[3];
    const float* W2   = (const float*)d_in[4];
    const float* b2   = (const float*)d_in[5];
    const float* Wout = (const float*)d_in[6];
    const float* bout = (const float*)d_in[7];
    float* out = (float*)d_out;

    const int N = in_sizes[0] / 128;
    const int E = in_sizes[1] / 2;
    const int* row = ei;        // edge_index[0] : scatter destination (segment id)
    const int* col = ei + E;    // edge_index[1] : gather source

    char* ws = (char*)d_ws;
    const size_t NB = (size_t)N * 128 * sizeof(float);
    float* agg = (float*)(ws);                  // reused for both aggregations
    float* x1  = (float*)(ws + NB);
    float* x2  = (float*)(ws + 2 * NB);
    float* deg = (float*)(ws + 3 * NB);         // degree -> inv_deg in place

    const int nThreads  = 256;
    const int nBlocksN  = (N + nThreads - 1) / nThreads;
    const int nBlocksE  = (E + nThreads - 1) / nThreads;
    const int n4        = N * 32;                              // float4 count
    const int copyBlks  = (n4 + nThreads - 1) / nThreads;
    const long scatterT = (long)E * 32;
    const int scatBlks  = (int)((scatterT + nThreads - 1) / nThreads);
    const int gemmBlks  = (N + 15) / 16;

    // degrees (edges + self loop), then invert
    fill_ones_kernel  <<<nBlocksN, nThreads, 0, stream>>>(deg, N);
    deg_edges_kernel  <<<nBlocksE, nThreads, 0, stream>>>(row, deg, E);
    invert_deg_kernel <<<nBlocksN, nThreads, 0, stream>>>(deg, N);

    // hop 1: agg = x (self) + scatter(x) ; x1 = relu(inv_deg * (agg @ W1) + b1)
    // GEMM1 dual-writes x1 and agg (fused self-loop init for hop 2).
    copy_f4_kernel      <<<copyBlks, nThreads, 0, stream>>>(x, agg, n4);
    scatter_edges_kernel<<<scatBlks, nThreads, 0, stream>>>(x, agg, row, col, E);
    gemm_wmma_kernel    <<<gemmBlks, nThreads, 0, stream>>>(agg, W1, nullptr, nullptr,
                                                            b1, deg, x1, agg, N, 1);

    // hop 2 over x1 (agg already initialized to x1 by GEMM1 epilogue)
    scatter_edges_kernel<<<scatBlks, nThreads, 0, stream>>>(x1, agg, row, col, E);
    gemm_wmma_kernel    <<<gemmBlks, nThreads, 0, stream>>>(agg, W2, nullptr, nullptr,
                                                            b2, deg, x2, nullptr, N, 1);

    // output: [x1 | x2] @ Wout + bout  (two K=128 passes, Wout split in halves)
    gemm_wmma_kernel    <<<gemmBlks, nThreads, 0, stream>>>(x1, Wout, x2, Wout + 128 * 128,
                                                            bout, nullptr, out, nullptr, N, 0);
}